// VectorQuantizer_5153960755464
// MI455X (gfx1250) — compile-verified
//
#include <hip/hip_runtime.h>
#include <cfloat>
#include <math.h>

typedef float v2f __attribute__((ext_vector_type(2)));
typedef float v4f __attribute__((ext_vector_type(4)));
typedef float v8f __attribute__((ext_vector_type(8)));

#define N_ROWS 16384
#define DIM    256
#define K_CODES 8192
#define COMMIT 0.25f

#if __has_builtin(__builtin_amdgcn_global_load_async_to_lds_b128) && \
    __has_builtin(__builtin_amdgcn_s_wait_asynccnt)
#define VQ_ASYNC 1
#else
#define VQ_ASYNC 0
#endif

// The async-to-LDS builtin takes pointers to a 16-byte int vector in
// AS(1) (global) / AS(3) (LDS), per the clang prototype.
typedef int vq_i32x4 __attribute__((vector_size(16)));
typedef __attribute__((address_space(1))) vq_i32x4* vq_g4p;  // global
typedef __attribute__((address_space(3))) vq_i32x4* vq_l4p;  // LDS

// Global pointer -> AS(1): numeric identity for global memory on AMDGCN
// (also drops const). LDS pointer -> AS(3): proper addrspacecast so the
// flat-aperture bits are stripped.
#define AS1V4(p) ((vq_g4p)(unsigned long long)(p))
#define AS3V4(p) ((vq_l4p)(p))

// ws layout (float units):
// [0, 8192)       eNorm
// [8192, 16384)   counts
// [16384, 32768)  indices (int)
// [32768]         errSum

__global__ __launch_bounds__(256) void vq_enorm_init(const float* __restrict__ emb,
                                                     float* __restrict__ eNorm,
                                                     float* __restrict__ counts,
                                                     float* __restrict__ errSum) {
    const int wave = threadIdx.x >> 5;
    const int lane = threadIdx.x & 31;
    const int row  = blockIdx.x * 8 + wave;
    const float* e = emb + row * DIM;
    float s = 0.f;
#pragma unroll
    for (int i = 0; i < 2; ++i) {
        v4f v = *(const v4f*)(e + i * 128 + lane * 4);
        s += v.x * v.x + v.y * v.y + v.z * v.z + v.w * v.w;
    }
#pragma unroll
    for (int m = 16; m >= 1; m >>= 1) s += __shfl_xor(s, m, 32);
    if (lane == 0) { eNorm[row] = s; counts[row] = 0.f; }
    if (blockIdx.x == 0 && threadIdx.x == 0) errSum[0] = 0.f;
}

// 4 waves per block; each wave owns 16 rows of X (A kept in registers).
// Codebook streamed as 16-code tiles, DMA'd global->LDS with the CDNA5
// async-to-LDS path (ASYNCcnt), double-buffered.
__global__ __launch_bounds__(128) void vq_argmin(const float* __restrict__ inputs,
                                                 const float* __restrict__ emb,
                                                 const float* __restrict__ eNorm,
                                                 int* __restrict__ outIdx) {
    __shared__ float Btile[2][16 * DIM];  // 2 x 16KB

    const int tid   = threadIdx.x;
    const int lane  = tid & 31;
    const int wave  = tid >> 5;
    const int nsub  = lane & 15;          // row (A) / col (B) within 16
    const int khalf = (lane >> 4) << 1;   // K offset within each 4-chunk: 0 or 2

    const int rowBase = blockIdx.x * 64 + wave * 16;

    // A fragments: row (rowBase + nsub), K elements {4j+khalf, 4j+khalf+1}
    v2f areg[64];
    {
        const float* ap = inputs + (rowBase + nsub) * DIM + khalf;
#pragma unroll
        for (int j = 0; j < 64; ++j) areg[j] = *(const v2f*)(ap + 4 * j);
    }

    float bestVal[8];
    int   bestIdx[8];
#pragma unroll
    for (int v = 0; v < 8; ++v) { bestVal[v] = FLT_MAX; bestIdx[v] = 0; }

    const int NT = K_CODES / 16;  // 512 tiles

#if VQ_ASYNC
    // Each tile is a flat 16KB slab of the codebook (16 consecutive rows).
    // 128 threads x 8 chunks x 16B = 16KB; 8 async instructions per wave.
    auto issue_tile = [&](int t, int buf) {
#pragma unroll
        for (int i = 0; i < 8; ++i) {
            const int q = tid + 128 * i;  // 16B chunk index within tile
            __builtin_amdgcn_global_load_async_to_lds_b128(
                AS1V4((const char*)emb + (size_t)t * 16 * DIM * 4 + (size_t)q * 16),
                AS3V4((char*)&Btile[buf][0] + q * 16),
                0, 0);
        }
    };

    issue_tile(0, 0);
    for (int t = 0; t < NT; ++t) {
        const int buf = t & 1;
        if (t + 1 < NT) {
            issue_tile(t + 1, buf ^ 1);               // prefetch next tile
            __builtin_amdgcn_s_wait_asynccnt(8);      // tile t's 8 DMAs done
        } else {
            __builtin_amdgcn_s_wait_asynccnt(0);
        }
        __syncthreads();  // all waves' DMAs for tile t visible in LDS

        v8f c = {0.f, 0.f, 0.f, 0.f, 0.f, 0.f, 0.f, 0.f};
        const float* bp = &Btile[buf][nsub * DIM + khalf];
#pragma unroll
        for (int j = 0; j < 64; ++j) {
            v2f b = *(const v2f*)(bp + 4 * j);
            c = __builtin_amdgcn_wmma_f32_16x16x4_f32(
                    false, areg[j], false, b, (short)0, c, false, false);
        }

        const float en   = eNorm[t * 16 + nsub];
        const int   code = t * 16 + nsub;
#pragma unroll
        for (int v = 0; v < 8; ++v) {
            float dist = en - 2.f * c[v];
            if (dist < bestVal[v]) { bestVal[v] = dist; bestIdx[v] = code; }
        }
        __syncthreads();  // everyone done reading buf before it is re-filled
    }
#else
    // Fallback: register-staged double buffering.
    v4f breg[8];
    auto stage_load = [&](int t) {
        const float* src = emb + t * 16 * DIM;
#pragma unroll
        for (int i = 0; i < 8; ++i) {
            int q = tid + 128 * i;
            breg[i] = *(const v4f*)(src + (q >> 6) * DIM + (q & 63) * 4);
        }
    };
    auto stage_store = [&](int buf) {
#pragma unroll
        for (int i = 0; i < 8; ++i) {
            int q = tid + 128 * i;
            *(v4f*)(&Btile[buf][q * 4]) = breg[i];
        }
    };

    stage_load(0);
    stage_store(0);
    __syncthreads();

    for (int t = 0; t < NT; ++t) {
        const int buf = t & 1;
        if (t + 1 < NT) stage_load(t + 1);

        v8f c = {0.f, 0.f, 0.f, 0.f, 0.f, 0.f, 0.f, 0.f};
        const float* bp = &Btile[buf][nsub * DIM + khalf];
#pragma unroll
        for (int j = 0; j < 64; ++j) {
            v2f b = *(const v2f*)(bp + 4 * j);
            c = __builtin_amdgcn_wmma_f32_16x16x4_f32(
                    false, areg[j], false, b, (short)0, c, false, false);
        }

        const float en   = eNorm[t * 16 + nsub];
        const int   code = t * 16 + nsub;
#pragma unroll
        for (int v = 0; v < 8; ++v) {
            float dist = en - 2.f * c[v];
            if (dist < bestVal[v]) { bestVal[v] = dist; bestIdx[v] = code; }
        }

        if (t + 1 < NT) {
            __syncthreads();
            stage_store(buf ^ 1);
        }
        __syncthreads();
    }
#endif

    // Cross-lane argmin within each 16-lane half (C layout: lanes 0-15 hold
    // M=v, lanes 16-31 hold M=v+8). Prefer lower code index on exact ties.
#pragma unroll
    for (int v = 0; v < 8; ++v) {
        float val = bestVal[v];
        int   idx = bestIdx[v];
#pragma unroll
        for (int m = 8; m >= 1; m >>= 1) {
            float ov = __shfl_xor(val, m, 32);
            int   oi = __shfl_xor(idx, m, 32);
            if (ov < val || (ov == val && oi < idx)) { val = ov; idx = oi; }
        }
        if (nsub == 0) {
            int row = rowBase + v + ((lane >> 4) << 3);
            outIdx[row] = idx;
        }
    }
}

__global__ __launch_bounds__(256) void vq_gather(const float* __restrict__ inputs,
                                                 const float* __restrict__ emb,
                                                 const int* __restrict__ idx,
                                                 float* __restrict__ outQ,
                                                 float* __restrict__ outIdxF,
                                                 float* __restrict__ counts,
                                                 float* __restrict__ errSum) {
    const int wave = threadIdx.x >> 5;
    const int lane = threadIdx.x & 31;
    const int row  = blockIdx.x * 8 + wave;
    const int k    = idx[row];
    const float* e = emb + k * DIM;
    const float* x = inputs + row * DIM;
    float* q = outQ + row * DIM;
    float s = 0.f;
#pragma unroll
    for (int i = 0; i < 2; ++i) {
        const int off = i * 128 + lane * 4;
        v4f ev = *(const v4f*)(e + off);
        v4f xv = *(const v4f*)(x + off);
        *(v4f*)(q + off) = ev;
        v4f d = ev - xv;
        s += d.x * d.x + d.y * d.y + d.z * d.z + d.w * d.w;
    }
#pragma unroll
    for (int m = 16; m >= 1; m >>= 1) s += __shfl_xor(s, m, 32);
    if (lane == 0) {
        atomicAdd(errSum, s);
        atomicAdd(&counts[k], 1.0f);
        outIdxF[row] = (float)k;
    }
}

__global__ __launch_bounds__(256) void vq_finalize(const float* __restrict__ counts,
                                                   const float* __restrict__ errSum,
                                                   float* __restrict__ outLoss,
                                                   float* __restrict__ outPpl) {
    __shared__ float red[8];
    const float invN = 1.0f / (float)N_ROWS;
    float local = 0.f;
    for (int k = threadIdx.x; k < K_CODES; k += 256) {
        float p = counts[k] * invN;
        local += p * logf(p + 1e-10f);
    }
#pragma unroll
    for (int m = 16; m >= 1; m >>= 1) local += __shfl_xor(local, m, 32);
    const int lane = threadIdx.x & 31, wave = threadIdx.x >> 5;
    if (lane == 0) red[wave] = local;
    __syncthreads();
    if (threadIdx.x == 0) {
        float tot = 0.f;
        for (int w = 0; w < 8; ++w) tot += red[w];
        *outPpl  = expf(-tot);
        // q_latent_loss is exactly 0; loss = 0.25 * mean((q - x)^2)
        *outLoss = COMMIT * errSum[0] / (float)(N_ROWS * DIM);
    }
}

extern "C" void kernel_launch(void* const* d_in, const int* in_sizes, int n_in,
                              void* d_out, int out_size, void* d_ws, size_t ws_size,
                              hipStream_t stream) {
    const float* inputs = (const float*)d_in[0];   // [16,1024,256] -> [16384,256]
    const float* emb    = (const float*)d_in[1];   // [8192,256]

    float* ws     = (float*)d_ws;
    float* eNorm  = ws;
    float* counts = ws + 8192;
    int*   idx    = (int*)(ws + 16384);
    float* errSum = ws + 32768;

    float* outQ    = (float*)d_out;                 // 4194304
    float* outIdxF = outQ + N_ROWS * DIM;           // 16384
    float* outLoss = outIdxF + N_ROWS;              // 1
    float* outPpl  = outLoss + 1;                   // 1

    vq_enorm_init<<<K_CODES / 8, 256, 0, stream>>>(emb, eNorm, counts, errSum);
    vq_argmin<<<N_ROWS / 64, 128, 0, stream>>>(inputs, emb, eNorm, idx);
    vq_gather<<<N_ROWS / 8, 256, 0, stream>>>(inputs, emb, idx, outQ, outIdxF, counts, errSum);
    vq_finalize<<<1, 256, 0, stream>>>(counts, errSum, outLoss, outPpl);
}